// MultiRelationSplitGNNLayer_90915867722266
// MI455X (gfx1250) — compile-verified
//
#include <hip/hip_runtime.h>
#include <math.h>

// ---------------------------------------------------------------------------
// Problem constants (from reference)
// ---------------------------------------------------------------------------
#define NN 100000
#define DD 64
#define OO 64
#define EE 1600000
#define RR 3
#define LRELU 0.01f

typedef float v2f __attribute__((ext_vector_type(2)));
typedef float v8f __attribute__((ext_vector_type(8)));

// ---------------------------------------------------------------------------
// GEMM: C[N x 64] (=|+=) act( A[N x 64] @ (c1*B1 + c2*B2)[64 x 64] + bias )
// flags bit0: accumulate into existing C;  bit1: leaky_relu at the end.
// One 256-thread block = 8 waves; each wave computes a 16x64 output tile
// with V_WMMA_F32_16X16X4_F32 (16 k-steps per 16-col tile).
//
// B is staged in LDS *pre-packed in K-pairs*:
//   Bs[((k>>1)*64 + col)*2 + (k&1)] = Bcombined[k][col]
// so each WMMA B fragment (B[4j+2h][col], B[4j+2h+1][col]) is one aligned
// ds_load_b64 into an even VGPR pair (WMMA operands must be even-aligned).
// ---------------------------------------------------------------------------
__global__ __launch_bounds__(256) void gemm64_kernel(
    const float* __restrict__ A, int nrows,
    const float* __restrict__ B1, float c1,
    const float* __restrict__ B2, float c2,
    const float* __restrict__ bias,
    float* __restrict__ C, int flags)
{
    __shared__ __align__(16) float Bs[64 * 64];   // packed layout, 16 KB
    __shared__ float bs[64];

    const int tid = threadIdx.x;
    // Stage combined B into LDS, packed in K-pairs
    for (int i = tid; i < 64 * 64; i += 256) {
        const int k = i >> 6, col = i & 63;
        float v = c1 * B1[i];
        if (B2) v += c2 * B2[i];
        Bs[((k >> 1) * 64 + col) * 2 + (k & 1)] = v;
    }
    if (tid < 64) bs[tid] = bias ? bias[tid] : 0.0f;
    __syncthreads();

    const int wave = tid >> 5;
    const int lane = tid & 31;
    const int half = lane >> 4;   // 0: lanes 0-15, 1: lanes 16-31
    const int l16  = lane & 15;
    const int r0   = blockIdx.x * 128 + wave * 16;

    // ---- A fragment: 16x64 rows r0..r0+15 -------------------------------
    // ISA 32-bit A 16x4 layout: lane<16 holds K=(4j,4j+1), lane>=16 K=(4j+2,4j+3)
    int arow = r0 + l16;
    if (arow >= nrows) arow = nrows - 1;          // clamp (stores predicated)
    const float* Ar = A + (size_t)arow * 64 + half * 2;
    v2f afrag[16];
#pragma unroll
    for (int j = 0; j < 16; ++j) {
        afrag[j].x = Ar[4 * j];
        afrag[j].y = Ar[4 * j + 1];
    }

    // ---- 4 column tiles of 16 ------------------------------------------
    v8f acc[4];
#pragma unroll
    for (int nt = 0; nt < 4; ++nt) {
        const float* Bp = Bs + (half * 64 + nt * 16 + l16) * 2;  // j=0 base
        v8f c = {};
#pragma unroll
        for (int j = 0; j < 16; ++j) {
            // packed pair index kp = 2j + half; float index = kp*128 + col*2
            const v2f b = *(const v2f*)(Bp + j * 256);
            c = __builtin_amdgcn_wmma_f32_16x16x4_f32(
                    false, afrag[j], false, b, (short)0, c, false, false);
        }
        acc[nt] = c;
    }

    // ---- store: C/D layout: VGPR v -> M = v + 8*half, col = nt*16 + l16 --
#pragma unroll
    for (int nt = 0; nt < 4; ++nt) {
#pragma unroll
        for (int v = 0; v < 8; ++v) {
            const int row = r0 + v + half * 8;
            if (row < nrows) {
                const int col = nt * 16 + l16;
                const size_t idx = (size_t)row * 64 + col;
                float val = acc[nt][v] + bs[col];
                if (flags & 1) val += C[idx];
                if (flags & 2) val = (val > 0.0f) ? val : val * LRELU;
                C[idx] = val;
            }
        }
    }
}

// ---------------------------------------------------------------------------
// Edge scoring: one wave per edge. pos[e] = (sp.(Wf0+Wf2) + dp.(Wf1-Wf2) + bf >= 0)
// (tanh is monotone with tanh(0)=0, so the sign of the linear score suffices)
// Also accumulates deg / pdeg / ndeg at dst via float atomics (counts are exact).
// ---------------------------------------------------------------------------
__global__ __launch_bounds__(256) void score_kernel(
    const float* __restrict__ proj, const float* __restrict__ Wf,
    const float* __restrict__ bf,
    const int* __restrict__ src, const int* __restrict__ dst,
    float* __restrict__ pos,
    float* __restrict__ deg, float* __restrict__ pdeg, float* __restrict__ ndeg,
    int E)
{
    __shared__ float w1[64], w2[64];
    const int tid = threadIdx.x;
    if (tid < 64) {
        w1[tid] = Wf[tid] + Wf[128 + tid];
        w2[tid] = Wf[64 + tid] - Wf[128 + tid];
    }
    __syncthreads();

    const int lane = tid & 31;
    const int gw = (blockIdx.x * 256 + tid) >> 5;
    const int nw = (gridDim.x * 256) >> 5;
    const float b = bf[0];

    for (int e = gw; e < E; e += nw) {
        const int s = src[e], d = dst[e];
        const float* sp = proj + (size_t)s * 64;
        const float* dp = proj + (size_t)d * 64;
        float part = sp[lane] * w1[lane] + sp[lane + 32] * w1[lane + 32]
                   + dp[lane] * w2[lane] + dp[lane + 32] * w2[lane + 32];
#pragma unroll
        for (int off = 16; off > 0; off >>= 1)
            part += __shfl_xor(part, off, 32);
        if (lane == 0) {
            const float p = (part + b) >= 0.0f ? 1.0f : 0.0f;
            pos[e] = p;
            atomicAdd(deg + d, 1.0f);
            atomicAdd(pdeg + d, p);
            atomicAdd(ndeg + d, 1.0f - p);
        }
    }
}

// deg -> max(deg,1)^-0.5 in place, for all three degree arrays
__global__ void dinv_kernel(float* deg, float* pdeg, float* ndeg, int n)
{
    const int i = blockIdx.x * blockDim.x + threadIdx.x;
    if (i < n) {
        deg[i]  = 1.0f / sqrtf(fmaxf(deg[i], 1.0f));
        pdeg[i] = 1.0f / sqrtf(fmaxf(pdeg[i], 1.0f));
        ndeg[i] = 1.0f / sqrtf(fmaxf(ndeg[i], 1.0f));
    }
}

__global__ void zero3_kernel(float* a, float* b, float* c, int n)
{
    const int i = blockIdx.x * blockDim.x + threadIdx.x;
    if (i < n) { a[i] = 0.0f; b[i] = 0.0f; c[i] = 0.0f; }
}

__global__ void zero_kernel(float* p, int n)
{
    const int i = blockIdx.x * blockDim.x + threadIdx.x;
    if (i < n) p[i] = 0.0f;
}

// out[n,i] = b_out[i]
__global__ void init_out_kernel(float* __restrict__ out, const float* __restrict__ b, int total)
{
    const int i = blockIdx.x * blockDim.x + threadIdx.x;
    if (i < total) out[i] = b[i & 63];
}

// ---------------------------------------------------------------------------
// Laplacian scatter: agg[dst] += (Dinv[src] * mask_e) * f[src]
// mode 0: mask=1, 1: mask=pos[e], 2: mask=1-pos[e]. One wave per edge; each
// lane moves 2 contiguous floats via global_atomic_add_f32. The next strided
// edge's source row is software-prefetched (global_prefetch_b8) to overlap
// the index-dependent gather latency.
// ---------------------------------------------------------------------------
__global__ __launch_bounds__(256) void prop_kernel(
    const float* __restrict__ f, const float* __restrict__ Dinv,
    const float* __restrict__ pos, int mode,
    const int* __restrict__ src, const int* __restrict__ dst,
    float* __restrict__ agg, int E)
{
    const int tid = threadIdx.x;
    const int lane = tid & 31;
    const int gw = (blockIdx.x * 256 + tid) >> 5;
    const int nw = (gridDim.x * 256) >> 5;

    for (int e = gw; e < E; e += nw) {
        const int en = e + nw;
        if (en < E) {
            const int sn = src[en];
            __builtin_prefetch(f + (size_t)sn * 64 + 2 * lane, 0, 0);
        }
        float m = 1.0f;
        if (mode == 1) m = pos[e];
        else if (mode == 2) m = 1.0f - pos[e];
        if (m == 0.0f) continue;            // wave-uniform, safe
        const int s = src[e], d = dst[e];
        const float cc = Dinv[s] * m;
        const float2 v = *(const float2*)(f + (size_t)s * 64 + 2 * lane);
        float* a = agg + (size_t)d * 64 + 2 * lane;
        atomicAdd(a, cc * v.x);
        atomicAdd(a + 1, cc * v.y);
    }
}

// f_next = f - agg * Dinv[row];  agg re-zeroed for the next step.
__global__ __launch_bounds__(256) void combine_kernel(
    const float* __restrict__ f, float* __restrict__ agg,
    const float* __restrict__ Dinv, float* __restrict__ fn, int n)
{
    const int t = blockIdx.x * blockDim.x + threadIdx.x;   // one float4 each
    if (t >= n * 16) return;
    const int row = t >> 4;
    const float di = Dinv[row];
    const float4 fv = ((const float4*)f)[t];
    const float4 av = ((float4*)agg)[t];
    float4 r;
    r.x = fv.x - av.x * di; r.y = fv.y - av.y * di;
    r.z = fv.z - av.z * di; r.w = fv.w - av.w * di;
    ((float4*)fn)[t] = r;
    ((float4*)agg)[t] = make_float4(0.f, 0.f, 0.f, 0.f);
}

// ---------------------------------------------------------------------------
// Host-side orchestration
// ---------------------------------------------------------------------------
extern "C" void kernel_launch(void* const* d_in, const int* in_sizes, int n_in,
                              void* d_out, int out_size, void* d_ws, size_t ws_size,
                              hipStream_t stream)
{
    (void)in_sizes; (void)n_in; (void)out_size; (void)ws_size;
    const int N = NN, E = EE, R = RR;

    const float* feat    = (const float*)d_in[0];
    const float* Wd      = (const float*)d_in[1];
    const float* bd      = (const float*)d_in[2];
    const float* Wf      = (const float*)d_in[3];
    const float* bf      = (const float*)d_in[4];
    const float* W_lin   = (const float*)d_in[5];   // (R,256,64)
    const float* b_lin   = (const float*)d_in[6];   // (R,64)
    const float* W_lin1  = (const float*)d_in[7];   // (R,256,64)
    const float* b_lin1  = (const float*)d_in[8];   // (R,64)
    const float* W_trans = (const float*)d_in[9];   // (R,64,64)
    const float* b_trans = (const float*)d_in[10];  // (R,64)
    const float* W_out   = (const float*)d_in[11];  // (576,64)
    const float* b_out   = (const float*)d_in[12];  // (64)
    const int*   src     = (const int*)d_in[13];    // (R,E)
    const int*   dst     = (const int*)d_in[14];    // (R,E)
    float* out = (float*)d_out;

    // workspace carve (~136 MB)
    float* ws   = (float*)d_ws;
    float* proj = ws; ws += (size_t)N * 64;
    float* fA   = ws; ws += (size_t)N * 64;
    float* fB   = ws; ws += (size_t)N * 64;
    float* agg  = ws; ws += (size_t)N * 64;
    float* hacc = ws; ws += (size_t)N * 64;
    float* posb = ws; ws += (size_t)E;
    float* deg  = ws; ws += N;
    float* pdeg = ws; ws += N;
    float* ndeg = ws; ws += N;

    const float TH[4][3] = {
        {1.0f, -0.5f, 0.25f}, {0.5f, 0.3f, -0.2f},
        {0.2f, -0.1f, 0.05f}, {0.8f, 0.4f, 0.1f}};

    const int GB  = (N + 127) / 128;          // gemm blocks
    const int NB  = (N + 255) / 256;
    const int CB  = (N * 16 + 255) / 256;     // combine blocks
    const int OB  = (N * 64 + 255) / 256;
    const int EGB = 4096;                     // edge-kernel grid (wave-strided)

    auto gemm = [&](const float* A, const float* B1, float c1, const float* B2,
                    float c2, const float* bias, float* C, int flags) {
        gemm64_kernel<<<GB, 256, 0, stream>>>(A, N, B1, c1, B2, c2, bias, C, flags);
    };

    // Runs a theta-branch: nf filters (theta rows th0..), 2*nf Laplacian hops.
    // Step j uses f_j; filter t contributes TH[th0+t][j-2t]*Wbase_block(blockOff+t).
    auto branch = [&](int nf, int th0, const float* Wbase, int blockOff,
                      const float* bias, const float* Dinv, int mode,
                      const int* sr, const int* dr, bool accFirst, bool actLast) {
        const float* fcur = feat;
        const int steps = 2 * nf;
        for (int j = 0; j <= steps; ++j) {
            const float* B1 = nullptr; const float* B2 = nullptr;
            float c1 = 0.f, c2 = 0.f;
            for (int t = 0; t < nf; ++t) {
                const int k = j - 2 * t;
                if (k >= 0 && k <= 2) {
                    const float* Bp = Wbase + (size_t)(blockOff + t) * 4096;
                    const float cc = TH[th0 + t][k];
                    if (!B1) { B1 = Bp; c1 = cc; } else { B2 = Bp; c2 = cc; }
                }
            }
            const bool last = (j == steps);
            const int flags = ((j == 0 && !accFirst) ? 0 : 1) |
                              ((last && actLast) ? 2 : 0);
            gemm(fcur, B1, c1, B2, c2, (last && actLast) ? bias : nullptr, hacc, flags);
            if (!last) {
                float* fn = (fcur == fA) ? fB : fA;
                prop_kernel<<<EGB, 256, 0, stream>>>(fcur, Dinv, posb, mode, sr, dr, agg, E);
                combine_kernel<<<CB, 256, 0, stream>>>(fcur, agg, Dinv, fn, N);
                fcur = fn;
            }
        }
    };

    // proj = feat @ Wd + bd
    gemm(feat, Wd, 1.0f, nullptr, 0.0f, bd, proj, 0);
    // out = b_out (broadcast); agg = 0 (then kept zeroed by combine_kernel)
    init_out_kernel<<<OB, 256, 0, stream>>>(out, b_out, N * 64);
    zero_kernel<<<OB, 256, 0, stream>>>(agg, N * 64);

    for (int r = 0; r < R; ++r) {
        const int* sr = src + (size_t)r * E;
        const int* dr = dst + (size_t)r * E;

        zero3_kernel<<<NB, 256, 0, stream>>>(deg, pdeg, ndeg, N);
        score_kernel<<<EGB, 256, 0, stream>>>(proj, Wf, bf, sr, dr, posb,
                                              deg, pdeg, ndeg, E);
        dinv_kernel<<<NB, 256, 0, stream>>>(deg, pdeg, ndeg, N);

        // ---- hs_o branch: 4 filters, full adjacency, Dinv = Di ----
        branch(4, 0, W_lin + (size_t)r * 256 * 64, 0, b_lin + r * 64,
               deg, 0, sr, dr, /*accFirst=*/false, /*actLast=*/true);
        gemm(hacc, W_out + (size_t)(0 + r) * 4096, 1.0f, nullptr, 0.0f,
             nullptr, out, 1);

        // ---- hs_p (filters 0-1, pos mask, Dp) then hs_n (filters 2-3, neg, Dn)
        branch(2, 0, W_lin1 + (size_t)r * 256 * 64, 0, nullptr,
               pdeg, 1, sr, dr, /*accFirst=*/false, /*actLast=*/false);
        branch(2, 2, W_lin1 + (size_t)r * 256 * 64, 2, b_lin1 + r * 64,
               ndeg, 2, sr, dr, /*accFirst=*/true, /*actLast=*/true);
        gemm(hacc, W_out + (size_t)(3 + r) * 4096, 1.0f, nullptr, 0.0f,
             nullptr, out, 1);

        // ---- tr = feat @ W_trans[r] + b_trans[r] ----
        gemm(feat, W_trans + (size_t)r * 4096, 1.0f, nullptr, 0.0f,
             b_trans + r * 64, hacc, 0);
        gemm(hacc, W_out + (size_t)(6 + r) * 4096, 1.0f, nullptr, 0.0f,
             nullptr, out, 1);
    }
}